// GcnNet_56478819942509
// MI455X (gfx1250) — compile-verified
//
#include <hip/hip_runtime.h>

// Problem constants (match reference)
#define NN       50000
#define NE       800000
#define DIN      256
#define DH       128
#define NLAB     1000

typedef __attribute__((ext_vector_type(2))) float v2f;
typedef __attribute__((ext_vector_type(8))) float v8f;

// Native fp32 WMMA: D = A(16x4) * B(4x16) + C(16x16)
__device__ __forceinline__ v8f wmma_k4(v2f a, v2f b, v8f c) {
  return __builtin_amdgcn_wmma_f32_16x16x4_f32(
      /*neg_a=*/false, a, /*neg_b=*/false, b,
      /*c_mod=*/(short)0, c, /*reuse_a=*/false, /*reuse_b=*/false);
}

// ---------------------------------------------------------------------------
// GEMM 1: h = relu(x @ W_in + b_in)   [NN x DIN] @ [DIN x DH]
// One wave32 per 16x16 output tile; tileN-major wave ordering for W reuse.
// ---------------------------------------------------------------------------
__global__ __launch_bounds__(256) void k_gemm_in(
    const float* __restrict__ x, const float* __restrict__ W,
    const float* __restrict__ bias, float* __restrict__ h) {
  const int wave  = (blockIdx.x * blockDim.x + threadIdx.x) >> 5;
  const int lane  = threadIdx.x & 31;
  const int tilesM = NN / 16;          // 3125
  const int tilesN = DH / 16;          // 8
  if (wave >= tilesM * tilesN) return;
  const int tileN = wave / tilesM;
  const int tileM = wave % tilesM;
  const int m0 = tileM * 16, n0 = tileN * 16;
  const int half = lane >> 4, l = lane & 15;

  // A lane base: row m0+l, K offset 0/2 by lane half (ISA 16x4 f32 layout)
  const float* arow = x + (size_t)(m0 + l) * DIN + 2 * half;
  // B lane base: col n0+l, row offset 0/2 by lane half
  const float* bcol = W + (size_t)(2 * half) * DH + (n0 + l);

  v8f acc = {};
#pragma unroll 8
  for (int k0 = 0; k0 < DIN; k0 += 4) {
    v2f a; a.x = arow[k0];               a.y = arow[k0 + 1];
    v2f b; b.x = bcol[(size_t)k0 * DH];  b.y = bcol[(size_t)(k0 + 1) * DH];
    acc = wmma_k4(a, b, acc);
  }
  const float bn = bias[n0 + l];
#pragma unroll
  for (int i = 0; i < 8; ++i) {
    const int row = m0 + i + 8 * half;
    float v = acc[i] + bn;
    h[(size_t)row * DH + (n0 + l)] = v > 0.f ? v : 0.f;
  }
}

// ---------------------------------------------------------------------------
// Zero res/agg accumulators (float4 grid-stride)
// ---------------------------------------------------------------------------
__global__ void k_zero(float4* __restrict__ p, long n4) {
  long i = blockIdx.x * (long)blockDim.x + threadIdx.x;
  const long stride = (long)gridDim.x * blockDim.x;
  const float4 z = make_float4(0.f, 0.f, 0.f, 0.f);
  for (; i < n4; i += stride) p[i] = z;
}

// ---------------------------------------------------------------------------
// Edge kernel: one wave per edge. Wave loads the 512B source row coalesced
// (float4/lane), scales by self_w and ppi_w, scatters with f32 HW atomics.
// ---------------------------------------------------------------------------
__global__ __launch_bounds__(256) void k_edge(
    const float* __restrict__ h, const int* __restrict__ src,
    const int* __restrict__ dst, const float* __restrict__ sw,
    const float* __restrict__ pw, float* __restrict__ res,
    float* __restrict__ agg) {
  const int e = (blockIdx.x * blockDim.x + threadIdx.x) >> 5;
  if (e >= NE) return;
  const int lane = threadIdx.x & 31;
  const int s = src[e], d = dst[e];
  const float ws = sw[e], wp = pw[e];
  const float4 v = *(const float4*)(h + (size_t)s * DH + lane * 4);
  float* rp = res + (size_t)d * DH + lane * 4;
  float* ap = agg + (size_t)d * DH + lane * 4;
  unsafeAtomicAdd(rp + 0, v.x * ws);
  unsafeAtomicAdd(rp + 1, v.y * ws);
  unsafeAtomicAdd(rp + 2, v.z * ws);
  unsafeAtomicAdd(rp + 3, v.w * ws);
  unsafeAtomicAdd(ap + 0, v.x * wp);
  unsafeAtomicAdd(ap + 1, v.y * wp);
  unsafeAtomicAdd(ap + 2, v.z * wp);
  unsafeAtomicAdd(ap + 3, v.w * wp);
}

// ---------------------------------------------------------------------------
// Layer GEMM: h = relu(agg @ W + b) + res    [NN x DH] @ [DH x DH]
// Writes h in place (old h is dead; kernel reads only agg/res).
// ---------------------------------------------------------------------------
__global__ __launch_bounds__(256) void k_gemm_layer(
    const float* __restrict__ agg, const float* __restrict__ W,
    const float* __restrict__ bias, const float* __restrict__ res,
    float* __restrict__ h) {
  const int wave  = (blockIdx.x * blockDim.x + threadIdx.x) >> 5;
  const int lane  = threadIdx.x & 31;
  const int tilesM = NN / 16;
  const int tilesN = DH / 16;
  if (wave >= tilesM * tilesN) return;
  const int tileN = wave / tilesM;
  const int tileM = wave % tilesM;
  const int m0 = tileM * 16, n0 = tileN * 16;
  const int half = lane >> 4, l = lane & 15;

  const float* arow = agg + (size_t)(m0 + l) * DH + 2 * half;
  const float* bcol = W + (size_t)(2 * half) * DH + (n0 + l);

  v8f acc = {};
#pragma unroll 8
  for (int k0 = 0; k0 < DH; k0 += 4) {
    v2f a; a.x = arow[k0];               a.y = arow[k0 + 1];
    v2f b; b.x = bcol[(size_t)k0 * DH];  b.y = bcol[(size_t)(k0 + 1) * DH];
    acc = wmma_k4(a, b, acc);
  }
  const float bn = bias[n0 + l];
#pragma unroll
  for (int i = 0; i < 8; ++i) {
    const int row = m0 + i + 8 * half;
    float v = acc[i] + bn;
    v = v > 0.f ? v : 0.f;
    h[(size_t)row * DH + (n0 + l)] = v + res[(size_t)row * DH + (n0 + l)];
  }
}

// ---------------------------------------------------------------------------
// Output GEMM: out = h @ W_out + b_out   [NN x DH] @ [DH x NLAB], NLAB=1000
// Column guard for the ragged last tile (63 tiles of 16 cover 1000).
// ---------------------------------------------------------------------------
__global__ __launch_bounds__(256) void k_gemm_out(
    const float* __restrict__ h, const float* __restrict__ W,
    const float* __restrict__ bias, float* __restrict__ out) {
  const int wave  = (blockIdx.x * blockDim.x + threadIdx.x) >> 5;
  const int lane  = threadIdx.x & 31;
  const int tilesM = NN / 16;                 // 3125
  const int tilesN = (NLAB + 15) / 16;        // 63
  if (wave >= tilesM * tilesN) return;
  const int tileN = wave / tilesM;
  const int tileM = wave % tilesM;
  const int m0 = tileM * 16, n0 = tileN * 16;
  const int half = lane >> 4, l = lane & 15;
  const int col = n0 + l;
  const bool colok = col < NLAB;

  const float* arow = h + (size_t)(m0 + l) * DH + 2 * half;
  const float* bcol = W + (size_t)(2 * half) * NLAB + col;

  v8f acc = {};
#pragma unroll 8
  for (int k0 = 0; k0 < DH; k0 += 4) {
    v2f a; a.x = arow[k0];  a.y = arow[k0 + 1];
    v2f b;
    b.x = colok ? bcol[(size_t)k0 * NLAB] : 0.f;
    b.y = colok ? bcol[(size_t)(k0 + 1) * NLAB] : 0.f;
    acc = wmma_k4(a, b, acc);
  }
  if (colok) {
    const float bn = bias[col];
#pragma unroll
    for (int i = 0; i < 8; ++i) {
      const int row = m0 + i + 8 * half;
      out[(size_t)row * NLAB + col] = acc[i] + bn;
    }
  }
}

// ---------------------------------------------------------------------------
extern "C" void kernel_launch(void* const* d_in, const int* in_sizes, int n_in,
                              void* d_out, int out_size, void* d_ws, size_t ws_size,
                              hipStream_t stream) {
  const float* x      = (const float*)d_in[0];
  const int*   src    = (const int*)d_in[1];
  const int*   dst    = (const int*)d_in[2];
  const float* self_w = (const float*)d_in[3];
  const float* ppi_w  = (const float*)d_in[4];
  const float* W_in   = (const float*)d_in[5];
  const float* b_in   = (const float*)d_in[6];
  const float* W1     = (const float*)d_in[7];
  const float* b1     = (const float*)d_in[8];
  const float* W2     = (const float*)d_in[9];
  const float* b2     = (const float*)d_in[10];
  const float* W_out  = (const float*)d_in[11];
  const float* b_out  = (const float*)d_in[12];
  float* out = (float*)d_out;

  // Workspace layout: h | res | agg  (each NN*DH floats = 25.6 MB)
  float* h   = (float*)d_ws;
  float* res = h + (size_t)NN * DH;
  float* agg = res + (size_t)NN * DH;

  const int tilesMN   = (NN / 16) * (DH / 16);          // 25000 waves
  const int gemmBlks  = (tilesMN + 7) / 8;              // 8 waves/block
  const int tilesOut  = (NN / 16) * ((NLAB + 15) / 16); // 196875 waves
  const int outBlks   = (tilesOut + 7) / 8;
  const int edgeBlks  = (NE * 32 + 255) / 256;          // one wave per edge
  const long zero4    = (long)2 * NN * DH / 4;          // res+agg in float4

  // 1) h = relu(x @ W_in + b_in)
  k_gemm_in<<<gemmBlks, 256, 0, stream>>>(x, W_in, b_in, h);

  // 2) two GCN layers
  const float* Ws[2] = {W1, W2};
  const float* bs[2] = {b1, b2};
  for (int layer = 0; layer < 2; ++layer) {
    k_zero<<<12500, 256, 0, stream>>>((float4*)res, zero4);
    k_edge<<<edgeBlks, 256, 0, stream>>>(h, src, dst, self_w, ppi_w, res, agg);
    k_gemm_layer<<<gemmBlks, 256, 0, stream>>>(agg, Ws[layer], bs[layer], res, h);
  }

  // 3) out = h @ W_out + b_out
  k_gemm_out<<<outBlks, 256, 0, stream>>>(h, W_out, b_out, out);
}